// ARMAGen_46608985096244
// MI455X (gfx1250) — compile-verified
//
#include <hip/hip_runtime.h>

// ---------------------------------------------------------------------------
// ARMA(4,4) generator, blocked-scan formulation on CDNA5 WMMA + async-to-LDS.
//
//   x[t] = mu + sum_i phi_i x[t-i] + eps[t] + sum_j theta_j eps[t-j]
//
// Per 16-step block with carried histories (x[t0-1..-4], eps[t0-1..-4]):
//   X(16n x 16t) = E * G (4 K-chunks) + Xh * CX + Eh * CE + mu*s[i]
// -> 6x V_WMMA_F32_16X16X4_F32 per (d, n-group) tile per time block.
// eps staging: GLOBAL_LOAD_ASYNC_TO_LDS_B128 (ASYNCcnt), double-buffered,
// issued one block ahead so HBM latency overlaps the WMMA chain.
// Barriers are hand-rolled (DScnt/ASYNCcnt only) so the per-block
// global_store_b128 drains in the background (no s_wait_storecnt in loop).
// ---------------------------------------------------------------------------

typedef float v2f __attribute__((ext_vector_type(2)));
typedef float v8f __attribute__((ext_vector_type(8)));
typedef int   v4i __attribute__((ext_vector_type(4)));

#define Nn 256
#define Tt 4096
#define Dd 64
#define TB 16            // time block
#define NBLK (Tt / TB)   // 256
#define DG 8             // d's per workgroup (one wave per d)
#define TAB_STRIDE 176   // floats per d: gpad[32] ms[16] cx[64] ce[64]

// eps LDS: per-(t,n) slot of 12 floats (48 B): 8 d-values + 4 pad.
//  - slot base 48B-aligned -> both 16B async b128 writes aligned
//  - A-operand reads hit banks 12*n mod 64 (permutation for n=0..15)
#define ESLOT 12
#define E2(buf, t, n, dl) ((((buf) * 256 + (t) * 16 + (n)) * ESLOT) + (dl))
// staged X tiles: [d][t][n] padded to 17
#define XIDX(dl, t, n) ((((dl) * TB + (t)) * 17) + (n))

#define AS1 __attribute__((address_space(1)))
#define AS3 __attribute__((address_space(3)))

__device__ __forceinline__ void async_b128(const float* g, float* l) {
    __builtin_amdgcn_global_load_async_to_lds_b128(
        (AS1 v4i*)(g), (AS3 v4i*)(l), 0, 0);
}

// Barrier ordering LDS traffic only (no STOREcnt drain).
__device__ __forceinline__ void bar_lds() {
    asm volatile("s_wait_dscnt 0x0\n\t"
                 "s_barrier_signal -1\n\t"
                 "s_barrier_wait -1" ::: "memory");
}
// Barrier publishing async-to-LDS arrivals + LDS traffic.
__device__ __forceinline__ void bar_publish() {
    asm volatile("s_wait_asynccnt 0x0\n\t"
                 "s_wait_dscnt 0x0\n\t"
                 "s_barrier_signal -1\n\t"
                 "s_barrier_wait -1" ::: "memory");
}

// --------------------------- setup: per-d tables ---------------------------
__global__ void arma_setup(const float* __restrict__ phi,
                           const float* __restrict__ theta,
                           const float* __restrict__ mu,
                           float* __restrict__ tab) {
    int d = threadIdx.x;
    if (d >= Dd) return;
    float ph[5], th[5];
    ph[0] = 0.f; th[0] = 0.f;
    for (int m = 1; m <= 4; ++m) {
        ph[m] = phi[d * 4 + m - 1];
        th[m] = theta[d * 4 + m - 1];
    }
    const float mud = mu[d];
    float h[16];
    h[0] = 1.f;
    for (int k = 1; k < 16; ++k) {
        float a = 0.f;
        for (int m = 1; m <= 4; ++m)
            if (k - m >= 0) a += ph[m] * h[k - m];
        h[k] = a;
    }
    float* t0 = tab + d * TAB_STRIDE;
    for (int i = 0; i < 16; ++i) t0[i] = 0.f;
    for (int k = 0; k < 16; ++k) {
        float g = h[k];
        for (int m = 1; m <= 4; ++m)
            if (k - m >= 0) g += th[m] * h[k - m];
        t0[16 + k] = g;
    }
    float s = 0.f;
    for (int i = 0; i < 16; ++i) { s += h[i]; t0[32 + i] = mud * s; }
    for (int l = 1; l <= 4; ++l)
        for (int i = 0; i < 16; ++i) {
            float cx = 0.f, ce = 0.f;
            for (int m = l; m <= 4; ++m) {
                int idx = i - (m - l);
                if (idx >= 0) { cx += ph[m] * h[idx]; ce += th[m] * h[idx]; }
            }
            t0[48  + (l - 1) * 16 + i] = cx;
            t0[112 + (l - 1) * 16 + i] = ce;
        }
}

// ------------------------------ main kernel --------------------------------
// grid = 128 blocks (16 n-groups x 8 d-groups), 256 threads (8 waves).
__global__ __launch_bounds__(256) void arma_main(const float* __restrict__ eps,
                                                 const float* __restrict__ x0,
                                                 const float* __restrict__ tab,
                                                 float* __restrict__ out) {
    __shared__ float ebuf[2 * 256 * ESLOT];  // double-buffered eps tiles (24 KB)
    __shared__ float xs[DG * TB * 17];       // staged X tiles (8.5 KB)

    const int tid  = threadIdx.x;
    const int ngrp = blockIdx.x & 15;
    const int dgrp = blockIdx.x >> 4;
    const int n0   = ngrp * 16;
    const int d0   = dgrp * DG;

    const int w  = tid >> 5;   // wave id == local d
    const int l  = tid & 31;
    const int li = l & 15;     // M/N index within operand
    const int hi = l >> 4;     // lane-half
    const int K  = hi << 1;    // K slot base: K_total = v + 2*hi
    const int d  = d0 + w;

    // (n,t) role for async staging + coalesced store phases
    const int pn = tid >> 4;   // 0..15
    const int pt = tid & 15;   // 0..15

    // ---- per-tile constants: B operands (layout N=l%16, K=v+2*hi) ----
    const float* tb = tab + d * TAB_STRIDE;
    v2f bg[4], bcx, bce;
#pragma unroll
    for (int j = 0; j < 4; ++j)
#pragma unroll
        for (int v = 0; v < 2; ++v)
            bg[j][v] = tb[16 + li - 4 * j - (v + K)];   // g[i - 4j - k], 0-padded
#pragma unroll
    for (int v = 0; v < 2; ++v) {
        bcx[v] = tb[48  + (v + K) * 16 + li];
        bce[v] = tb[112 + (v + K) * 16 + li];
    }
    const float msval = tb[32 + li];   // mu * s_i
    v8f cms;
#pragma unroll
    for (int v = 0; v < 8; ++v) cms[v] = msval;

    // ---- initial histories (A layout: M=l%16, K=v+2*hi) ----
    v2f xh, eh;
    xh[0] = (hi == 0) ? x0[(n0 + li) * Dd + d] : 0.f;  // k=0 -> x[-1]
    xh[1] = 0.f;
    eh[0] = 0.f;
    eh[1] = 0.f;

    const float* gload = eps + (size_t)(n0 + pn) * Tt * Dd + d0;
    float*       gstor = out + (size_t)(n0 + pn) * Tt * Dd + d0;

    // ---- prologue: async-stage block 0 into buffer 0 ----
    {
        const float* p = gload + (size_t)pt * Dd;
        float* lq = &ebuf[E2(0, pt, pn, 0)];
        async_b128(p, lq);
        async_b128(p + 4, lq + 4);
    }
    bar_publish();

    for (int b = 0; b < NBLK; ++b) {
        const int cur = b & 1;

        // ---- issue next block's async staging (overlaps with compute) ----
        if (b + 1 < NBLK) {
            const float* p = gload + (size_t)((b + 1) * TB + pt) * Dd;
            float* lq = &ebuf[E2(1 - cur, pt, pn, 0)];
            async_b128(p, lq);
            async_b128(p + 4, lq + 4);
        }

        // ---- per-wave tile compute: 6x v_wmma_f32_16x16x4_f32 ----
        v2f a0, a1, a2, a3;
#pragma unroll
        for (int v = 0; v < 2; ++v) {
            a0[v] = ebuf[E2(cur,  0 + v + K, li, w)];
            a1[v] = ebuf[E2(cur,  4 + v + K, li, w)];
            a2[v] = ebuf[E2(cur,  8 + v + K, li, w)];
            a3[v] = ebuf[E2(cur, 12 + v + K, li, w)];
        }
        v8f acc;
        acc = __builtin_amdgcn_wmma_f32_16x16x4_f32(false, a0, false, bg[0],
                                                    (short)0, cms, false, false);
        acc = __builtin_amdgcn_wmma_f32_16x16x4_f32(false, a1, false, bg[1],
                                                    (short)0, acc, false, false);
        acc = __builtin_amdgcn_wmma_f32_16x16x4_f32(false, a2, false, bg[2],
                                                    (short)0, acc, false, false);
        acc = __builtin_amdgcn_wmma_f32_16x16x4_f32(false, a3, false, bg[3],
                                                    (short)0, acc, false, false);
        acc = __builtin_amdgcn_wmma_f32_16x16x4_f32(false, xh, false, bcx,
                                                    (short)0, acc, false, false);
        acc = __builtin_amdgcn_wmma_f32_16x16x4_f32(false, eh, false, bce,
                                                    (short)0, acc, false, false);

        // D[n,i]: lane holds N=i=li, rows M=n=v+8*hi -> stage as xs[d][t][n]
#pragma unroll
        for (int v = 0; v < 8; ++v)
            xs[XIDX(w, li, v + 8 * hi)] = acc[v];

        // next-block histories (same-wave LDS, DS ops in-order per wave)
#pragma unroll
        for (int v = 0; v < 2; ++v) {
            const int k = v + K;
            xh[v] = xs[XIDX(w, 15 - k, li)];
            eh[v] = ebuf[E2(cur, 15 - k, li, w)];
        }
        bar_lds();   // xs complete across all waves (DScnt only)

        // ---- coalesced global store of X block (drains in background) ----
        {
            float4 o0, o1;
            o0.x = xs[XIDX(0, pt, pn)];
            o0.y = xs[XIDX(1, pt, pn)];
            o0.z = xs[XIDX(2, pt, pn)];
            o0.w = xs[XIDX(3, pt, pn)];
            o1.x = xs[XIDX(4, pt, pn)];
            o1.y = xs[XIDX(5, pt, pn)];
            o1.z = xs[XIDX(6, pt, pn)];
            o1.w = xs[XIDX(7, pt, pn)];
            float* q = gstor + (size_t)(b * TB + pt) * Dd;
            *(float4*)q       = o0;
            *(float4*)(q + 4) = o1;
        }

        // ---- publish next eps buffer; also fences xs reuse ----
        bar_publish();
    }
}

// ------------------------------- launcher ----------------------------------
extern "C" void kernel_launch(void* const* d_in, const int* in_sizes, int n_in,
                              void* d_out, int out_size, void* d_ws, size_t ws_size,
                              hipStream_t stream) {
    const float* eps   = (const float*)d_in[0];
    const float* phi   = (const float*)d_in[1];
    const float* theta = (const float*)d_in[2];
    const float* mu    = (const float*)d_in[3];
    const float* x0    = (const float*)d_in[4];
    float* out = (float*)d_out;
    float* tab = (float*)d_ws;   // 176 * 64 * 4 B = 45 KB

    arma_setup<<<1, 64, 0, stream>>>(phi, theta, mu, tab);
    arma_main<<<dim3((Dd / DG) * 16), 256, 0, stream>>>(eps, x0, tab, out);
}